// Subtraction_48009144435373
// MI455X (gfx1250) — compile-verified
//
#include <hip/hip_runtime.h>
#include <stdint.h>

// SAN "subtraction" op: out[b,c, ki*7+kj, h*56+w] = x[b,c,h,w] - xp[b,c,h+ki,w+kj]
// x: [8,64,56,56] f32, reflect pad 3, 7x7, stride 1, dilation 1.
// Store-bandwidth bound (321 MB moved, 78.7 MFLOP). Strategy:
//   * TDM tensor_load_to_lds stages each 12.5KB plane into LDS (one DMA slice per wave)
//   * branchless reflect indexing from LDS
//   * coalesced 128-bit non-temporal global stores

typedef float    f32x4 __attribute__((ext_vector_type(4)));
typedef uint32_t u32x4 __attribute__((ext_vector_type(4)));
typedef uint32_t u32x8 __attribute__((ext_vector_type(8)));

#define SAN_H   56
#define SAN_W   56
#define SAN_HW  (SAN_H * SAN_W)      // 3136
#define SAN_K   7
#define SAN_KK  (SAN_K * SAN_K)      // 49
#define SAN_PAD 3
#define WAVES_PER_BLOCK 8
#define ROWS_PER_WAVE   7            // 56 rows / 8 waves
#define SLICE_ELEMS     (ROWS_PER_WAVE * SAN_W)  // 392 floats per wave DMA

__device__ __forceinline__ int refl55(int d) {
    // reflect index for size-56 axis, valid for d in [-3, 58]
    int a = d < 0 ? -d : d;
    int b = 55 - a;
    b = b < 0 ? -b : b;
    return 55 - b;
}

__global__ __launch_bounds__(256) void san_sub_kernel(const float* __restrict__ x,
                                                      float* __restrict__ out) {
    __shared__ __align__(16) float smem[SAN_HW];

    const int plane = blockIdx.x;          // b*64 + c, 0..511
    const int tid   = threadIdx.x;

    // ---- Stage plane into LDS via Tensor Data Mover (one slice per wave32) ----
    {
        const uint32_t wave  = (uint32_t)tid >> 5;          // wave-uniform
        const uint32_t slice = wave * (uint32_t)SLICE_ELEMS;

        const uint64_t ga = (uint64_t)(uintptr_t)(x + (size_t)plane * SAN_HW + slice);
        const uint32_t lds_off =
            (uint32_t)(uintptr_t)(void*)smem + slice * 4u;  // low 32 bits = LDS byte offset

        const uint32_t L = (uint32_t)SLICE_ELEMS;           // 392 elements, contiguous

        // D# group 0 (128b): count=1 | lds_addr | global_addr[56:0] | type=2
        u32x4 g0;
        g0.x = 1u;                                          // count=1 (valid user D#)
        g0.y = lds_off;                                     // lds_addr (bytes)
        g0.z = (uint32_t)ga;                                // global_addr[31:0]
        g0.w = (uint32_t)((ga >> 32) & 0x01FFFFFFu) | (2u << 30); // addr[56:32] | type=2

        // D# group 1 (256b): data_size=4B, 1-D tile of L elements
        u32x8 g1;
        g1[0] = 2u << 16;                       // wg_mask=0 (not in cluster), data_size=2 (4B)
        g1[1] = (L & 0xFFFFu) << 16;            // tensor_dim0[15:0]  @ bits[63:48]
        g1[2] = (L >> 16) | (1u << 16);         // tensor_dim0[31:16] | tensor_dim1=1
        g1[3] = (L & 0xFFFFu) << 16;            // tensor_dim1[31:16]=0 | tile_dim0=L
        g1[4] = 1u;                             // tile_dim1=1 | tile_dim2=0
        g1[5] = L;                              // tensor_dim0_stride[31:0]
        g1[6] = 0u;                             // stride0[47:32] | dim1_stride[15:0]
        g1[7] = 0u;                             // dim1_stride[47:16]

        // 2-group form: tensors up to 2D (VADDR2/VADDR3 = NULL)
        asm volatile("tensor_load_to_lds %0, %1" :: "s"(g0), "s"(g1) : "memory");
    }

#if __has_builtin(__builtin_amdgcn_s_wait_tensorcnt)
    __builtin_amdgcn_s_wait_tensorcnt(0);
#else
    asm volatile("s_wait_tensorcnt 0x0" ::: "memory");
#endif
    __syncthreads();   // all 8 slices visible to all waves

    // ---- Compute + store: each thread owns float4 groups of hw positions ----
    const size_t out_plane = (size_t)plane * (size_t)SAN_KK * (size_t)SAN_HW;

    for (int g = tid; g < SAN_HW / 4; g += 256) {           // 784 groups
        const int h  = g / (SAN_W / 4);                     // row
        const int w0 = (g % (SAN_W / 4)) * 4;               // first col of float4

        const f32x4 c4 = *(const f32x4*)(smem + h * SAN_W + w0);

#pragma unroll
        for (int ki = 0; ki < SAN_K; ++ki) {
            const int rh = refl55(h + ki - SAN_PAD);
            const float* __restrict__ row = smem + rh * SAN_W;
#pragma unroll
            for (int kj = 0; kj < SAN_K; ++kj) {
                const int dw = w0 + kj - SAN_PAD;
                f32x4 o;
                o.x = c4.x - row[refl55(dw + 0)];
                o.y = c4.y - row[refl55(dw + 1)];
                o.z = c4.z - row[refl55(dw + 2)];
                o.w = c4.w - row[refl55(dw + 3)];

                f32x4* dst = (f32x4*)(out + out_plane
                                          + (size_t)(ki * SAN_K + kj) * SAN_HW
                                          + (size_t)g * 4u);
                __builtin_nontemporal_store(o, dst);        // global_store_b128 ... nt
            }
        }
    }
}

extern "C" void kernel_launch(void* const* d_in, const int* in_sizes, int n_in,
                              void* d_out, int out_size, void* d_ws, size_t ws_size,
                              hipStream_t stream) {
    const float* x  = (const float*)d_in[0];
    float* out      = (float*)d_out;

    dim3 grid(8 * 64);   // one workgroup per (b, c) plane
    dim3 block(256);     // 8 wave32s
    hipLaunchKernelGGL(san_sub_kernel, grid, block, 0, stream, x, out);
}